// GD_Block_57715770524142
// MI455X (gfx1250) — compile-verified
//
#include <hip/hip_runtime.h>

typedef __attribute__((ext_vector_type(2))) float v2f;
typedef __attribute__((ext_vector_type(8))) float v8f;

#define DIM 256
#define LDS_STRIDE 258   // 258 % 64 = 2 -> conflict-free 16-lane column reads; 258*4 % 8 == 0
#define MB 48            // rows per block tile (3 x 16 M-subtiles); LDS = 48*258*4 = 49,536 B

// ---------------------------------------------------------------- zero fill
__global__ __launch_bounds__(256) void zero_f32(float* __restrict__ p, long n) {
  long i = (long)blockIdx.x * blockDim.x + threadIdx.x;
  long stride = (long)gridDim.x * blockDim.x;
  for (; i < n; i += stride) p[i] = 0.0f;
}

// ------------------------------------------------- TAGConv scatter: agg[dst] += x[src]
// one thread per (edge, 4-channel quad): E*64 threads
__global__ __launch_bounds__(256) void tag_scatter(const float* __restrict__ x,
                                                   const int* __restrict__ src,
                                                   const int* __restrict__ dst,
                                                   float* __restrict__ agg, int E) {
  long idx = (long)blockIdx.x * blockDim.x + threadIdx.x;
  int e = (int)(idx >> 6);
  if (e >= E) return;
  int c4 = (int)(idx & 63) << 2;
  int s = src[e], d = dst[e];
  float4 val = *(const float4*)(x + (size_t)s * DIM + c4);
  float* p = agg + (size_t)d * DIM + c4;
  atomicAdd(p + 0, val.x);
  atomicAdd(p + 1, val.y);
  atomicAdd(p + 2, val.z);
  atomicAdd(p + 3, val.w);
}

// ------------------------------------------------- fp32 WMMA GEMM:
//   C[M,256] = A[M,256] @ W[256,256]            (A2 == nullptr)
//   C[M,256] = A@W + A2@W2                      (A2 != nullptr, fused dual-GEMM)
// block = 256 threads = 8 waves; block tile = 48 rows x 256 cols.
// wave w owns N columns [32w, 32w+32) across all three 16-row M-subtiles:
//   per K-step: 4 global B loads, 3 LDS A loads, 6 v_wmma_f32_16x16x4_f32.
__global__ __launch_bounds__(256) void gemm_wmma(const float* __restrict__ A,
                                                 const float* __restrict__ W,
                                                 const float* __restrict__ A2,
                                                 const float* __restrict__ W2,
                                                 float* __restrict__ C, int M) {
  __shared__ float As[MB * LDS_STRIDE];
  const int tid  = threadIdx.x;
  const int row0 = blockIdx.x * MB;
  const int wave = tid >> 5;
  const int lane = tid & 31;
  const int hi   = lane >> 4;   // lane group: 0 -> K pair {0,1}, 1 -> K pair {2,3}
  const int nl   = lane & 15;   // column within a 16-wide N-tile
  const int n0   = wave * 32;   // this wave's two N-tiles

  v8f acc[3][2] = {};

  const float* Asrc = A;
  const float* Wsrc = W;
  for (int pass = 0; pass < 2; ++pass) {
    // cooperative coalesced load of the MB x 256 A tile into LDS (zero-pad past M)
    for (int i = tid; i < MB * DIM; i += 256) {
      int r = i >> 8;
      int c = i & 255;
      int gr = row0 + r;
      As[r * LDS_STRIDE + c] = (gr < M) ? Asrc[(size_t)gr * DIM + c] : 0.0f;
    }
    __syncthreads();

    for (int kk = 0; kk < DIM; kk += 4) {
      const int kc = kk + 2 * hi;
      // B 4x16 fragments (row-striped per ISA): VGPR0 = W[kc][n], VGPR1 = W[kc+1][n]
      v2f b0, b1;
      b0.x = Wsrc[(size_t)kc * DIM + n0 + nl];
      b0.y = Wsrc[(size_t)(kc + 1) * DIM + n0 + nl];
      b1.x = Wsrc[(size_t)kc * DIM + n0 + 16 + nl];
      b1.y = Wsrc[(size_t)(kc + 1) * DIM + n0 + 16 + nl];
#pragma unroll
      for (int s = 0; s < 3; ++s) {
        // A 16x4 fragment: lane m = lane&15; VGPR0 = A[m][kc], VGPR1 = A[m][kc+1]
        v2f a = *(const v2f*)(&As[(s * 16 + (lane & 15)) * LDS_STRIDE + kc]);
        acc[s][0] = __builtin_amdgcn_wmma_f32_16x16x4_f32(false, a, false, b0, (short)0, acc[s][0], false, false);
        acc[s][1] = __builtin_amdgcn_wmma_f32_16x16x4_f32(false, a, false, b1, (short)0, acc[s][1], false, false);
      }
    }

    if (!A2) break;          // uniform branch
    __syncthreads();         // all waves done reading LDS before pass-2 reload
    Asrc = A2;
    Wsrc = W2;
  }

  // C layout: VGPR r holds rows r (lanes 0-15) / r+8 (lanes 16-31)
#pragma unroll
  for (int s = 0; s < 3; ++s) {
#pragma unroll
    for (int r = 0; r < 8; ++r) {
      int gr = row0 + s * 16 + r + 8 * hi;
      if (gr < M) {
        C[(size_t)gr * DIM + n0 + nl]      = acc[s][0][r];
        C[(size_t)gr * DIM + n0 + 16 + nl] = acc[s][1][r];
      }
    }
  }
}

// ------------------------------------------------- attention branch, fused subtract:
// per edge e: a = dot(q[d],k[s]); out[d] -= a * v[s].  One wave32 per edge.
__global__ __launch_bounds__(256) void attn_edge(const float* __restrict__ q,
                                                 const float* __restrict__ k,
                                                 const float* __restrict__ v,
                                                 const int* __restrict__ s2,
                                                 const int* __restrict__ d2,
                                                 float* __restrict__ out, int E) {
  int gw   = (int)(((long)blockIdx.x * blockDim.x + threadIdx.x) >> 5);
  int lane = threadIdx.x & 31;
  if (gw >= E) return;
  int s = s2[gw], d = d2[gw];

  const float4* qd = (const float4*)(q + (size_t)d * DIM);
  const float4* ks = (const float4*)(k + (size_t)s * DIM);
  float4 qa = qd[lane], qb = qd[lane + 32];
  float4 ka = ks[lane], kb = ks[lane + 32];
  float p = qa.x * ka.x + qa.y * ka.y + qa.z * ka.z + qa.w * ka.w
          + qb.x * kb.x + qb.y * kb.y + qb.z * kb.z + qb.w * kb.w;
#pragma unroll
  for (int off = 16; off > 0; off >>= 1) p += __shfl_xor(p, off, 32);

  const float4* vs = (const float4*)(v + (size_t)s * DIM);
  float4 va = vs[lane], vb = vs[lane + 32];
  float* od = out + (size_t)d * DIM;
  atomicAdd(od + 4 * lane + 0, -p * va.x);
  atomicAdd(od + 4 * lane + 1, -p * va.y);
  atomicAdd(od + 4 * lane + 2, -p * va.z);
  atomicAdd(od + 4 * lane + 3, -p * va.w);
  atomicAdd(od + 128 + 4 * lane + 0, -p * vb.x);
  atomicAdd(od + 128 + 4 * lane + 1, -p * vb.y);
  atomicAdd(od + 128 + 4 * lane + 2, -p * vb.z);
  atomicAdd(od + 128 + 4 * lane + 3, -p * vb.w);
}

extern "C" void kernel_launch(void* const* d_in, const int* in_sizes, int n_in,
                              void* d_out, int out_size, void* d_ws, size_t ws_size,
                              hipStream_t stream) {
  const float* x  = (const float*)d_in[0];
  const int* ei1  = (const int*)d_in[1];
  const int* ei2  = (const int*)d_in[2];
  const float* W0 = (const float*)d_in[3];
  const float* W1 = (const float*)d_in[4];
  const float* Wq = (const float*)d_in[5];
  const float* Wk = (const float*)d_in[6];
  const float* Wv = (const float*)d_in[7];
  float* out = (float*)d_out;

  const int N  = in_sizes[0] / DIM;
  const int E1 = in_sizes[1] / 2;
  const int E2 = in_sizes[2] / 2;
  const size_t NC = (size_t)N * DIM;

  // workspace layout: [buf0 | buf1 | buf2], each N*256 f32.
  // buf0 serves as `agg` first, then is recycled as `q`.
  float* agg = (float*)d_ws;
  float* qb  = agg;
  float* kb  = agg + NC;
  float* vb  = agg + 2 * NC;

  // 1) agg = 0
  zero_f32<<<2048, 256, 0, stream>>>(agg, (long)NC);

  // 2) agg[dst] += x[src]   (TAGConv hop-1 aggregation)
  {
    long threads = (long)E1 * 64;
    int blocks = (int)((threads + 255) / 256);
    tag_scatter<<<blocks, 256, 0, stream>>>(x, ei1, ei1 + E1, agg, E1);
  }

  const int gblocks = (N + MB - 1) / MB;
  // 3) out = x @ W0 + agg @ W1   (fused dual-GEMM, no C read-modify-write pass)
  gemm_wmma<<<gblocks, 256, 0, stream>>>(x, W0, agg, W1, out, N);

  // 4) k, v, then q (q reuses agg's buffer — safe: agg no longer read after step 3)
  gemm_wmma<<<gblocks, 256, 0, stream>>>(x, Wk, nullptr, nullptr, kb, N);
  gemm_wmma<<<gblocks, 256, 0, stream>>>(x, Wv, nullptr, nullptr, vb, N);
  gemm_wmma<<<gblocks, 256, 0, stream>>>(x, Wq, nullptr, nullptr, qb, N);

  // 5) out -= segment_sum(attn * v[src])
  {
    long threads = (long)E2 * 32;
    int blocks = (int)((threads + 255) / 256);
    attn_edge<<<blocks, 256, 0, stream>>>(qb, kb, vb, ei2, ei2 + E2, out, E2);
  }
}